// DTI_predictor_8031588843837
// MI455X (gfx1250) — compile-verified
//
#include <hip/hip_runtime.h>

#define N_MOL  2048
#define P_PRO  2048
#define HID    64
#define HEADS  16
#define NGRAPH 64

typedef __attribute__((ext_vector_type(2))) float v2f;
typedef __attribute__((ext_vector_type(8))) float v8f;

__device__ __forceinline__ float elu1(float x) {
    return x > 0.0f ? x : (expf(x) - 1.0f);
}

// ---------------------------------------------------------------------------
// Kernel 1: fused projection GEMMs via V_WMMA_F32_16X16X4_F32 (fp32 exact).
// One wave per 16-row tile; blocks 0..127 -> a_mol (+bmu), 128..255 -> a_pro.
// K=64 -> 16 chained WMMA ops accumulating the 16x16 (rows x heads) tile.
// ---------------------------------------------------------------------------
__global__ void proj_wmma(const float* __restrict__ mol,
                          const float* __restrict__ pro,
                          const float* __restrict__ Wmu,   // [2*HID, HEADS] row-major
                          const float* __restrict__ bmu,   // [HEADS]
                          float* __restrict__ a_mol,       // [N_MOL, HEADS]
                          float* __restrict__ a_pro) {     // [P_PRO, HEADS]
    const int  t      = blockIdx.x;               // 0..255
    const bool is_pro = (t >= N_MOL / 16);
    const int  tile   = is_pro ? (t - N_MOL / 16) : t;
    const int  lane   = threadIdx.x;              // wave32
    const int  half   = lane >> 4;                // 0: K lo pair, 1: K hi pair
    const int  l      = lane & 15;
    const int  row0   = tile * 16;

    const float* feats = is_pro ? pro : mol;
    const float* wb    = Wmu + (is_pro ? HID * HEADS : 0);

    v8f acc = {};
    const int arow = row0 + l;
#pragma unroll
    for (int k = 0; k < HID / 4; ++k) {
        const int kb = 4 * k + 2 * half;
        // A tile 16x4: lane l<16 -> row l, K=kb..kb+1 ; lane>=16 -> K=kb+2..kb+3
        const float2 av = *(const float2*)(feats + arow * HID + kb);
        v2f a; a.x = av.x; a.y = av.y;
        // B tile 4x16: b.x = Wmu[kb+0][l], b.y = Wmu[kb+1][l]
        v2f b;
        b.x = wb[(kb + 0) * HEADS + l];
        b.y = wb[(kb + 1) * HEADS + l];
        acc = __builtin_amdgcn_wmma_f32_16x16x4_f32(
            /*neg_a=*/false, a, /*neg_b=*/false, b,
            /*c_mod=*/(short)0, acc, /*reuse_a=*/false, /*reuse_b=*/false);
    }

    const float bias = is_pro ? 0.0f : bmu[l];    // N (head) == lane&15
    float* outp = is_pro ? a_pro : a_mol;
#pragma unroll
    for (int v = 0; v < 8; ++v) {
        const int r = row0 + half * 8 + v;        // C/D layout: M = v + 8*half
        outp[r * HEADS + l] = acc[v] + bias;
    }
}

// ---------------------------------------------------------------------------
// Kernel 2: zero the segment-sum accumulator (ws is not re-poisoned between
// replays, so we must re-zero every launch).
// ---------------------------------------------------------------------------
__global__ void zero_y(float* __restrict__ y) {
    const int idx = blockIdx.x * 256 + threadIdx.x;
    if (idx < NGRAPH * HEADS) y[idx] = 0.0f;
}

// ---------------------------------------------------------------------------
// Kernel 3: the heavy fused cross-sum.
//   y[batch[i]][h] += 2048 + sum_j elu(a_mol[i][h] + a_pro[j][h])
// 256 threads = 16 mol atoms x 16 heads; a_pro streamed through LDS in
// 128x16 (8 KB) chunks. Lanes 0-15 and 16-31 read identical LDS addresses
// (broadcast); the 16 heads hit 16 distinct banks -> conflict-free.
// ---------------------------------------------------------------------------
__global__ void cross_sum(const float* __restrict__ a_mol,
                          const float* __restrict__ a_pro,
                          const int* __restrict__ batch,
                          float* __restrict__ y) {
    __shared__ float sp[128 * HEADS];             // 8 KB chunk of a_pro
    const int tid  = threadIdx.x;                 // 0..255
    const int h    = tid & 15;
    const int isub = tid >> 4;
    const int i    = blockIdx.x * 16 + isub;

    const float am = a_mol[i * HEADS + h];
    float acc = 0.0f;

    for (int c = 0; c < P_PRO; c += 128) {
        const float4* src = (const float4*)(a_pro + c * HEADS);
        float4* dst = (float4*)sp;
#pragma unroll
        for (int u = tid; u < 128 * HEADS / 4; u += 256) dst[u] = src[u];
        __syncthreads();
#pragma unroll 8
        for (int j = 0; j < 128; ++j) {
            acc += elu1(am + sp[j * HEADS + h]);
        }
        __syncthreads();
    }

    acc += (float)P_PRO;                          // the "+1.0" summed over P
    atomicAdd(&y[batch[i] * HEADS + h], acc);     // sorted segments, few collisions
}

// ---------------------------------------------------------------------------
// Kernel 4: tail MLP: out[b] = elu(0.001*y[b] @ W1 + b1) @ W2 + b2
// One thread per graph; everything L0-resident.
// ---------------------------------------------------------------------------
__global__ void head_mlp(const float* __restrict__ y,
                         const float* __restrict__ W1,   // [HEADS, 2*HEADS]
                         const float* __restrict__ b1,   // [2*HEADS]
                         const float* __restrict__ W2,   // [2*HEADS]
                         const float* __restrict__ b2,   // [1]
                         float* __restrict__ out) {
    const int g = threadIdx.x;
    if (g >= NGRAPH) return;
    float yv[HEADS];
#pragma unroll
    for (int hh = 0; hh < HEADS; ++hh) yv[hh] = y[g * HEADS + hh] * 0.001f;
    float o = b2[0];
#pragma unroll
    for (int k = 0; k < 2 * HEADS; ++k) {
        float s = b1[k];
#pragma unroll
        for (int hh = 0; hh < HEADS; ++hh) s += yv[hh] * W1[hh * (2 * HEADS) + k];
        o += elu1(s) * W2[k];
    }
    out[g] = o;
}

extern "C" void kernel_launch(void* const* d_in, const int* in_sizes, int n_in,
                              void* d_out, int out_size, void* d_ws, size_t ws_size,
                              hipStream_t stream) {
    const float* mol   = (const float*)d_in[0];
    const float* pro   = (const float*)d_in[1];
    const float* Wmu   = (const float*)d_in[2];
    const float* bmu   = (const float*)d_in[3];
    const float* W1    = (const float*)d_in[4];
    const float* b1    = (const float*)d_in[5];
    const float* W2    = (const float*)d_in[6];
    const float* b2    = (const float*)d_in[7];
    const int*   batch = (const int*)d_in[8];     // jnp int64 -> int32 (x64 off)
    float* out = (float*)d_out;

    float* ws    = (float*)d_ws;
    float* a_mol = ws;                            // [N_MOL, HEADS]
    float* a_pro = ws + (size_t)N_MOL * HEADS;    // [P_PRO, HEADS]
    float* y     = ws + (size_t)(N_MOL + P_PRO) * HEADS; // [NGRAPH, HEADS]

    hipLaunchKernelGGL(zero_y, dim3(4), dim3(256), 0, stream, y);
    hipLaunchKernelGGL(proj_wmma, dim3((N_MOL + P_PRO) / 16), dim3(32), 0, stream,
                       mol, pro, Wmu, bmu, a_mol, a_pro);
    hipLaunchKernelGGL(cross_sum, dim3(N_MOL / 16), dim3(256), 0, stream,
                       a_mol, a_pro, batch, y);
    hipLaunchKernelGGL(head_mlp, dim3(1), dim3(64), 0, stream,
                       y, W1, b1, W2, b2, out);
}